// GATv2Layer_60533269070357
// MI455X (gfx1250) — compile-verified
//
#include <hip/hip_runtime.h>
#include <hip/hip_bf16.h>
#include <stdint.h>

// GATv2 layer for MI455X (gfx1250, wave32).
// Pipeline: bias -> proj GEMM (WMMA f32, TDM-staged A panel) -> masked softmax
//           -> AV batched GEMM (WMMA f32) -> out GEMM (WMMA f32, TDM-staged A)

typedef __attribute__((ext_vector_type(2))) float v2f;
typedef __attribute__((ext_vector_type(8))) float v8f;
typedef __attribute__((ext_vector_type(4))) unsigned int u32x4;
typedef __attribute__((ext_vector_type(4))) int i32x4;
typedef __attribute__((ext_vector_type(8))) int i32x8;

#define B_  4
#define N_  256
#define D_  256
#define H_  8
#define DH_ 32
#define NEG_SLOPE 0.2f

// workspace layout (float offsets)
#define OFF_X    0                         // [B*H, N, DH]   262144
#define OFF_BIAS (OFF_X + B_*H_*N_*DH_)    // [N, N]          65536
#define OFF_PT   (OFF_BIAS + N_*N_)        // [B*H, Nj, Ni] 2097152 (transposed, unnormalized)
#define OFF_INV  (OFF_PT + B_*H_*N_*N_)    // [B*H, N]         8192
#define OFF_Y    (OFF_INV + B_*H_*N_)      // [B*N, D]       262144

#if defined(__HIP_DEVICE_COMPILE__) && __has_builtin(__builtin_amdgcn_tensor_load_to_lds)
#define HAVE_TDM 1
#else
#define HAVE_TDM 0
#endif

#if HAVE_TDM
// Issue one TDM 2-D tile load (tileRows x tileCols f32, row stride strideElts)
// from global gsrc into LDS byte-offset ldsOff. D# layout per CDNA5 ISA ch.8:
//   group0: [1:0]=count=1 | lds_addr | global_addr[56:0] | type=2
//   group1: data_size=4B, tensor_dim0/1, tile_dim0/1, tensor_dim0_stride
//   groups 2/3 (+ trailing group): zero (2-D tensor; tile_dim2..4 = 0 => unused)
// This toolchain's builtin takes the 6-arg form:
//   (u32x4 g0, i32x8 g1, i32x4 g2, i32x4 g3, i32x8 g4, i32 cpol)
static __device__ inline void tdm_load_2d_f32(const float* gsrc, unsigned ldsOff,
                                              unsigned tileRows, unsigned tileCols,
                                              unsigned strideElts) {
    unsigned long long ga = (unsigned long long)(uintptr_t)gsrc;
    u32x4 g0;
    g0[0] = 1u;                                          // count=1, user D#
    g0[1] = ldsOff;                                      // lds_addr (bytes)
    g0[2] = (unsigned)(ga & 0xFFFFFFFFu);                // global_addr[31:0]
    g0[3] = (unsigned)((ga >> 32) & 0x01FFFFFFu)         // global_addr[56:32]
          | (2u << 30);                                  // type = 2 ("image")
    i32x8 g1;
    g1[0] = (int)(2u << 16);                             // data_size = 4 bytes
    g1[1] = (int)((tileCols & 0xFFFFu) << 16);           // tensor_dim0[15:0]
    g1[2] = (int)(((tileCols >> 16) & 0xFFFFu)           // tensor_dim0[31:16]
          | ((tileRows & 0xFFFFu) << 16));               // tensor_dim1[15:0]
    g1[3] = (int)(((tileRows >> 16) & 0xFFFFu)           // tensor_dim1[31:16]
          | ((tileCols & 0xFFFFu) << 16));               // tile_dim0
    g1[4] = (int)(tileRows & 0xFFFFu);                   // tile_dim1 (tile_dim2=0)
    g1[5] = (int)strideElts;                             // tensor_dim0_stride[31:0]
    g1[6] = 0;                                           // stride[47:32], d1_stride lo
    g1[7] = 0;
    i32x4 z4 = {0, 0, 0, 0};
    i32x8 z8 = {0, 0, 0, 0, 0, 0, 0, 0};
    __builtin_amdgcn_tensor_load_to_lds(g0, g1, z4, z4, z8, 0);
    __builtin_amdgcn_s_wait_tensorcnt(0);
}
#endif

// ---------------------------------------------------------------------------
// Kernel 0: fused mask + log-bias:  bias[i,j] = mask ? log(A0+1e-8) : -inf
// ---------------------------------------------------------------------------
__global__ void gat_bias_kernel(const uint8_t* __restrict__ mask,
                                const float* __restrict__ A0,
                                float* __restrict__ bias) {
    int i = blockIdx.x * blockDim.x + threadIdx.x;
    if (i < N_ * N_)
        bias[i] = mask[i] ? __logf(A0[i] + 1e-8f) : -__builtin_inff();
}

// ---------------------------------------------------------------------------
// 1024x256 @ 256x256 fp32 GEMM. Block = 4 waves; the block's 16x256 A row
// panel is TDM-loaded into LDS once and shared; each wave computes a 16x32
// C slab (two 16x16 accumulators off one A fragment), V_WMMA_F32_16X16X4_F32.
// Grid: 64 row panels x 2 column halves = 128 blocks.
// MODE 0: C[row, col] plain.   MODE 1: split heads -> X[b,h,n,dh].
// ---------------------------------------------------------------------------
template <int MODE>
__global__ __launch_bounds__(128) void gat_gemm_kernel(const float* __restrict__ A,
                                                       const float* __restrict__ Bm,
                                                       float* __restrict__ C) {
    __shared__ float as[16 * D_];                  // 16 KB A panel
    const int lane = threadIdx.x & 31;
    const int wave = threadIdx.x >> 5;
    const int pm = blockIdx.x >> 1;                // row panel (64)
    const int ph = blockIdx.x & 1;                 // column half (2)
    const int r0 = pm * 16;
    const int c0 = ph * 128 + wave * 32;           // this wave: cols c0..c0+31

    // ---- stage A[r0:r0+16, 0:256] into LDS ----
#if HAVE_TDM
    if (wave == 0) {
        tdm_load_2d_f32(A + (size_t)r0 * D_,
                        (unsigned)(uintptr_t)(void*)as, 16u, (unsigned)D_,
                        (unsigned)D_);
    }
#else
    for (int idx = threadIdx.x; idx < 16 * D_; idx += 128)
        as[idx] = A[(size_t)r0 * D_ + idx];
#endif
    __syncthreads();

    const int arow  = lane & 15;                   // local A row
    const int col0  = c0 + (lane & 15);            // tile 0 column
    const int col1  = col0 + 16;                   // tile 1 column
    const int khalf = (lane >> 4) * 2;             // lanes 16-31 hold K+2,K+3

    v8f acc0 = {}, acc1 = {};
    for (int k0 = 0; k0 < D_; k0 += 4) {
        const int kb = k0 + khalf;
        v2f a = *(const v2f*)(as + arow * D_ + kb);     // ds_load_b64, shared
        v2f b0, b1;
        b0.x = Bm[(kb + 0) * D_ + col0];
        b0.y = Bm[(kb + 1) * D_ + col0];
        b1.x = Bm[(kb + 0) * D_ + col1];
        b1.y = Bm[(kb + 1) * D_ + col1];
        acc0 = __builtin_amdgcn_wmma_f32_16x16x4_f32(false, a, false, b0,
                                                     (short)0, acc0, false, false);
        acc1 = __builtin_amdgcn_wmma_f32_16x16x4_f32(false, a, false, b1,
                                                     (short)0, acc1, false, false);
    }

    const int rbase = r0 + ((lane >> 4) << 3);     // C: VGPR v -> row rbase+v
#pragma unroll
    for (int v = 0; v < 8; ++v) {
        const int row = rbase + v;
        if (MODE == 0) {
            C[row * D_ + col0] = acc0[v];
            C[row * D_ + col1] = acc1[v];
        } else {
            const int b = row >> 8, n = row & 255;
            C[((b * H_ + (col0 >> 5)) * N_ + n) * DH_ + (col0 & 31)] = acc0[v];
            C[((b * H_ + (col1 >> 5)) * N_ + n) * DH_ + (col1 & 31)] = acc1[v];
        }
    }
}

// ---------------------------------------------------------------------------
// Kernel 2: GATv2 scores + masked softmax.  One block per (b,h), thread = row i.
// Head tile X[bh] (256x32 = 32KB) staged in LDS; inner j-loop reads LDS at a
// uniform address (broadcast). Two-pass online softmax (max, then exp+sum);
// writes UNNORMALIZED exp transposed Pt[j][i] (coalesced) and 1/sum per row,
// normalization folded into the AV GEMM.
// ---------------------------------------------------------------------------
__global__ __launch_bounds__(256) void gat_attn_kernel(const float* __restrict__ X,
                                                       const float* __restrict__ avec,
                                                       const float* __restrict__ bias,
                                                       float* __restrict__ Pt,
                                                       float* __restrict__ inv_sum) {
    __shared__ float xs[N_ * DH_];                 // 32 KB of the 320 KB WGP LDS
    const int bh  = blockIdx.x;
    const int h   = bh & (H_ - 1);
    const int tid = threadIdx.x;                   // row i

    __builtin_prefetch(bias + tid * N_, 0, 3);     // global_prefetch of bias row

    const float* Xh = X + (size_t)bh * N_ * DH_;
    for (int idx = tid; idx < N_ * DH_; idx += 256) xs[idx] = Xh[idx];
    __syncthreads();

    float av[DH_], xi[DH_];
#pragma unroll
    for (int d = 0; d < DH_; ++d) av[d] = avec[h * DH_ + d];
#pragma unroll
    for (int d = 0; d < DH_; ++d) xi[d] = xs[tid * DH_ + d];

    // pass 1: row max
    float m = -__builtin_inff();
    for (int j = 0; j < N_; ++j) {
        float s = 0.f;
#pragma unroll
        for (int d = 0; d < DH_; ++d) {
            const float t = xi[d] + xs[j * DH_ + d];
            s = __builtin_fmaf(av[d], (t > 0.f ? t : NEG_SLOPE * t), s);
        }
        m = fmaxf(m, s + bias[tid * N_ + j]);
    }

    // pass 2: exp, sum, transposed store
    float sum = 0.f;
    float* PtRow = Pt + (size_t)bh * N_ * N_;
    for (int j = 0; j < N_; ++j) {
        float s = 0.f;
#pragma unroll
        for (int d = 0; d < DH_; ++d) {
            const float t = xi[d] + xs[j * DH_ + d];
            s = __builtin_fmaf(av[d], (t > 0.f ? t : NEG_SLOPE * t), s);
        }
        const float p = __expf(s + bias[tid * N_ + j] - m);
        sum += p;
        PtRow[j * N_ + tid] = p;                   // coalesced across lanes
    }
    inv_sum[bh * N_ + tid] = 1.f / sum;
}

// ---------------------------------------------------------------------------
// Kernel 3: Y = softmax(P) @ X per (b,h), WMMA f32. A[i,k] = Pt[k,i]; the
// per-row 1/sum scale is applied on the accumulator. Writes merged [b,n,d].
// ---------------------------------------------------------------------------
__global__ __launch_bounds__(128) void gat_av_kernel(const float* __restrict__ Pt,
                                                     const float* __restrict__ X,
                                                     const float* __restrict__ inv_sum,
                                                     float* __restrict__ Y) {
    const int lane = threadIdx.x & 31;
    const int wave = threadIdx.x >> 5;
    const int tile = blockIdx.x * 4 + wave;        // 1024 tiles = 32 bh * (16x2)
    const int bh = tile >> 5;
    const int t  = tile & 31;
    const int tm = t >> 1, tn = t & 1;
    const int r0 = tm * 16, c0 = tn * 16;

    const float* P  = Pt + (size_t)bh * N_ * N_;   // [j][i]
    const float* Xh = X + (size_t)bh * N_ * DH_;   // [k][dh]

    const int irow  = r0 + (lane & 15);
    const int col   = c0 + (lane & 15);            // dh in 0..31
    const int khalf = (lane >> 4) * 2;

    v8f acc = {};
    for (int k0 = 0; k0 < N_; k0 += 4) {
        const int kb = k0 + khalf;
        v2f a;                                     // A[i,k] = Pt[k,i]
        a.x = P[(kb + 0) * N_ + irow];
        a.y = P[(kb + 1) * N_ + irow];
        v2f b;
        b.x = Xh[(kb + 0) * DH_ + col];
        b.y = Xh[(kb + 1) * DH_ + col];
        acc = __builtin_amdgcn_wmma_f32_16x16x4_f32(false, a, false, b,
                                                    (short)0, acc, false, false);
    }

    const int rbase = r0 + ((lane >> 4) << 3);
    const int b = bh >> 3, h = bh & (H_ - 1);
#pragma unroll
    for (int v = 0; v < 8; ++v) {
        const int n = rbase + v;
        const float val = acc[v] * inv_sum[bh * N_ + n];
        Y[((size_t)(b * N_ + n)) * D_ + h * DH_ + col] = val;
    }
}

// ---------------------------------------------------------------------------
extern "C" void kernel_launch(void* const* d_in, const int* in_sizes, int n_in,
                              void* d_out, int out_size, void* d_ws, size_t ws_size,
                              hipStream_t stream) {
    const float*   H_in  = (const float*)d_in[0];   // [B,N,D]
    const uint8_t* mask  = (const uint8_t*)d_in[1]; // [N,N] bool
    const float*   A0    = (const float*)d_in[2];   // [N,N]
    const float*   W_lin = (const float*)d_in[3];   // [D,D]
    const float*   avec  = (const float*)d_in[4];   // [H,DH]
    const float*   W_out = (const float*)d_in[5];   // [D,D]
    float*         out   = (float*)d_out;           // [B,N,D]

    float* wsf   = (float*)d_ws;
    float* Xw    = wsf + OFF_X;
    float* biasw = wsf + OFF_BIAS;
    float* Ptw   = wsf + OFF_PT;
    float* invw  = wsf + OFF_INV;
    float* Yw    = wsf + OFF_Y;

    // 0: bias = mask ? log(A0+1e-8) : -inf
    gat_bias_kernel<<<(N_ * N_ + 255) / 256, 256, 0, stream>>>(mask, A0, biasw);
    // 1: X = H @ W_lin, head-split layout (TDM-staged A panel, 2 tiles/wave)
    gat_gemm_kernel<1><<<128, 128, 0, stream>>>(H_in, W_lin, Xw);
    // 2: scores + masked softmax (one WGP workgroup per (b,h))
    gat_attn_kernel<<<B_ * H_, 256, 0, stream>>>(Xw, avec, biasw, Ptw, invw);
    // 3: Y = attn @ X, merge heads
    gat_av_kernel<<<256, 128, 0, stream>>>(Ptw, Xw, invw, Yw);
    // 4: out = Y @ W_out
    gat_gemm_kernel<0><<<128, 128, 0, stream>>>(Yw, W_out, out);
}